// SumRateLoss_9448928051420
// MI455X (gfx1250) — compile-verified
//
#include <hip/hip_runtime.h>

// CDNA5 (gfx1250) wave32 WMMA kernel for the sum-rate loss.
// Packs 2 batch items per 16x16 V_WMMA_F32_16X16X4_F32 tile, reduction m=64
// via 16 chained K=4 WMMA steps per accumulator. F32 WMMA has no A/B negate
// (NEG = {CNeg,0,0}), so we run 4 positive accumulator chains and combine:
//   re = sum(Hr*Vr) - sum(Hi*Vi),  im = sum(Hr*Vi) + sum(Hi*Vr).

typedef __attribute__((ext_vector_type(2))) float v2f;
typedef __attribute__((ext_vector_type(8))) float v8f;

#define WAVES_PER_BLOCK 8

__global__ __launch_bounds__(256) void sumrate_wmma_kernel(
    const float* __restrict__ Hr, const float* __restrict__ Hi,
    const float* __restrict__ Vr, const float* __restrict__ Vi,
    const float* __restrict__ noise_p, float* __restrict__ ws, int T)
{
    __shared__ float sred[WAVES_PER_BLOCK];
    const int lane = threadIdx.x & 31;
    const int wid  = threadIdx.x >> 5;
    const int tile = blockIdx.x * WAVES_PER_BLOCK + wid;

    float wavesum = 0.0f;
    if (tile < T) {
        // Per-lane invariant offset within this tile's 2-item slab (1024 floats).
        // A 16x4 fp32 layout: VGPR j <- K-row (lane<16 ? j : j+2), row = lane%16.
        // B 4x16 fp32 layout is symmetric (row pairing {j, j+2} across lane halves).
        // row/col % 8 -> k/l index, row/col / 8 -> which of the two batch items.
        const int k8    = lane & 7;          // k (A rows) == l (B cols) index
        const int nsel  = (lane >> 3) & 1;   // batch item within tile
        const int mhalf = lane >> 4;         // K-row pairing: {0,1} vs {2,3}
        const int inv   = k8 + nsel * 512 + mhalf * 16;
        const size_t base = (size_t)tile * 1024 + (size_t)inv;

        const float* hr = Hr + base;
        const float* hi = Hi + base;
        const float* vr = Vr + base;
        const float* vi = Vi + base;

        v8f accP = {};   // sum Hr*Vr
        v8f accQ = {};   // sum Hi*Vi
        v8f accR = {};   // sum Hr*Vi
        v8f accS = {};   // sum Hi*Vr
#pragma unroll
        for (int c = 0; c < 16; ++c) {       // m-chunks of 4
            const int o0 = 32 * c;           // +VGPR1 is +8 floats (next m row)
            const int o1 = 32 * c + 8;
            v2f Ar = { hr[o0], hr[o1] };
            v2f Ai = { hi[o0], hi[o1] };
            v2f Br = { vr[o0], vr[o1] };
            v2f Bi = { vi[o0], vi[o1] };
            accP = __builtin_amdgcn_wmma_f32_16x16x4_f32(
                false, Ar, false, Br, (short)0, accP, false, false);
            accQ = __builtin_amdgcn_wmma_f32_16x16x4_f32(
                false, Ai, false, Bi, (short)0, accQ, false, false);
            accR = __builtin_amdgcn_wmma_f32_16x16x4_f32(
                false, Ar, false, Bi, (short)0, accR, false, false);
            accS = __builtin_amdgcn_wmma_f32_16x16x4_f32(
                false, Ai, false, Br, (short)0, accS, false, false);
        }

        // |h_k^T v_l|^2 ; this lane holds column l = k8, rows k = VGPR index.
        float n2[8];
#pragma unroll
        for (int v = 0; v < 8; ++v) {
            const float re = accP[v] - accQ[v];
            const float im = accR[v] + accS[v];
            n2[v] = re * re + im * im;
        }

        // Row-sum over l: reduce across the 8-lane column group (stays inside
        // lanes 0-7 resp. 24-31 under xor 1/2/4).
        float S[8];
#pragma unroll
        for (int v = 0; v < 8; ++v) {
            float s = n2[v];
            s += __shfl_xor(s, 1, 32);
            s += __shfl_xor(s, 2, 32);
            s += __shfl_xor(s, 4, 32);
            S[v] = s;
        }

        // Lane with column index k8 owns the diagonal element (k = l = k8):
        // it computes the rate for user k8 of its batch item.
        float nom  = n2[0];
        float Ssel = S[0];
#pragma unroll
        for (int v = 1; v < 8; ++v) {
            nom  = (k8 == v) ? n2[v] : nom;
            Ssel = (k8 == v) ? S[v]  : Ssel;
        }
        const float noise = noise_p[0];
        const float denom = Ssel - nom + noise;
        const float rate  = __log2f(1.0f + nom / denom);

        // Valid output lanes: 0-7 (item n0 diag block) and 24-31 (item n1).
        const bool valid = (lane < 8) || (lane >= 24);
        float contrib = valid ? rate : 0.0f;
#pragma unroll
        for (int mask = 1; mask < 32; mask <<= 1)
            contrib += __shfl_xor(contrib, mask, 32);
        wavesum = contrib;
    }

    if (lane == 0) sred[wid] = wavesum;
    __syncthreads();
    if (threadIdx.x == 0) {
        float t = 0.0f;
#pragma unroll
        for (int i = 0; i < WAVES_PER_BLOCK; ++i) t += sred[i];
        ws[blockIdx.x] = t;
    }
}

// Deterministic final reduction of per-block partials; applies -1/N.
__global__ __launch_bounds__(256) void sumrate_final_reduce(
    const float* __restrict__ ws, int P, float scale, float* __restrict__ out)
{
    __shared__ float s[256];
    float acc = 0.0f;
    for (int i = threadIdx.x; i < P; i += 256) acc += ws[i];
    s[threadIdx.x] = acc;
    __syncthreads();
    for (int off = 128; off > 0; off >>= 1) {
        if ((int)threadIdx.x < off) s[threadIdx.x] += s[threadIdx.x + off];
        __syncthreads();
    }
    if (threadIdx.x == 0) out[0] = s[0] * scale;
}

extern "C" void kernel_launch(void* const* d_in, const int* in_sizes, int n_in,
                              void* d_out, int out_size, void* d_ws, size_t ws_size,
                              hipStream_t stream)
{
    const float* Hr    = (const float*)d_in[0];
    const float* Hi    = (const float*)d_in[1];
    const float* Vr    = (const float*)d_in[2];
    const float* Vi    = (const float*)d_in[3];
    const float* noise = (const float*)d_in[4];

    const int N = in_sizes[0] / 512;   // M*K = 64*8 floats per batch item
    const int T = N / 2;               // 2 batch items per 16x16 WMMA tile
    const int blocks = (T + WAVES_PER_BLOCK - 1) / WAVES_PER_BLOCK;

    float* ws = (float*)d_ws;
    sumrate_wmma_kernel<<<blocks, 256, 0, stream>>>(Hr, Hi, Vr, Vi, noise, ws, T);
    sumrate_final_reduce<<<1, 256, 0, stream>>>(ws, blocks, -1.0f / (float)N,
                                                (float*)d_out);
}